// SelectiveModel_77068893160206
// MI455X (gfx1250) — compile-verified
//
#include <hip/hip_runtime.h>
#include <hip/hip_bf16.h>

// ---------------------------------------------------------------------------
// SelectiveModel for MI455X (gfx1250), single fused kernel.
//
// Key algebraic facts exploited (see analysis):
//  * mem slots always hold exact embedding rows -> state = 8 token ids/batch
//  * pair@gw1 = ctxproj[tok] + memproj[slot_tok]  (both precomputable, 64 rows)
//  * sigmoid/gb2 monotone -> argmax on raw dot product
//  * q@rw1_top + rb1 -> qproj[66][64] precomputable
//  * remaining GEMMs (summary@rw1_bot, h@rw2) run on V_WMMA_F32_16X16X4_F32
//    in full f32 precision, 16-batch tiles per wave32.
// HBM traffic ~ seqs(3MB) + out(8MB) -> sub-microsecond at 23.3 TB/s.
// ---------------------------------------------------------------------------

#define B_TOTAL   32768
#define SEQ_LEN   24
#define T_STEPS   23      // SEQ_LEN - 1
#define H_DIM     64
#define SLOTS     8
#define VOCAB     64
#define NVOCAB    66      // VOCAB + 2 (query vocab)
#define G1        32      // gate hidden width
#define CP_STRIDE 36      // padded stride (floats) for 32-wide tables: 16B aligned, spreads LDS banks
#define E_STRIDE  68      // padded stride (floats) for 64-wide tables
#define THREADS   256     // 8 wave32s
#define WAVES     8

typedef float v2f __attribute__((ext_vector_type(2)));
typedef float v8f __attribute__((ext_vector_type(8)));

// D = A(16x4,f32) * B(4x16,f32) + C(16x16,f32)   -- CDNA5 VOP3P WMMA
__device__ __forceinline__ v8f wmma_f32_16x16x4(v2f a, v2f b, v8f c) {
#if defined(__HIP_DEVICE_COMPILE__) && __has_builtin(__builtin_amdgcn_wmma_f32_16x16x4_f32)
  // 8 args: (neg_a, A, neg_b, B, c_mod, C, reuse_a, reuse_b)
  return __builtin_amdgcn_wmma_f32_16x16x4_f32(false, a, false, b, (short)0, c, false, false);
#else
  // Host-pass placeholder only; never executed on gfx1250.
  c[0] += a[0] * b[0];
  return c;
#endif
}

__device__ __forceinline__ void wave_fence() {
#if defined(__HIP_DEVICE_COMPILE__) && __has_builtin(__builtin_amdgcn_wave_barrier)
  __builtin_amdgcn_wave_barrier();   // compiler fence; DS ops are in-order per wave in HW
#endif
}

__global__ __launch_bounds__(THREADS)
void selective_model_kernel(const int* __restrict__ seqs,
                            const int* __restrict__ qtok,
                            const float* __restrict__ embed,
                            const float* __restrict__ gw1,
                            const float* __restrict__ gb1,
                            const float* __restrict__ gw2,
                            const float* __restrict__ rw1,
                            const float* __restrict__ rb1,
                            const float* __restrict__ rw2,
                            const float* __restrict__ rb2,
                            float* __restrict__ out) {
  // ---- LDS (~90 KB of the 320 KB / WGP) ----
  __shared__ __align__(16) float embed_s[VOCAB * E_STRIDE];     // 17.4 KB
  __shared__ __align__(16) float ctxp_s [VOCAB * CP_STRIDE];    //  9.2 KB  (gb1 folded in)
  __shared__ __align__(16) float memp_s [VOCAB * CP_STRIDE];    //  9.2 KB
  __shared__ __align__(16) float qproj_s[NVOCAB * H_DIM];       // 16.9 KB  (rb1 folded in)
  __shared__ __align__(16) float h_s    [WAVES * 16 * E_STRIDE];// 34.8 KB  (per-wave h tile)
  __shared__ unsigned char slot_s[THREADS][SLOTS];              //  2.0 KB

  const int tid = threadIdx.x;
  const int blockBase = blockIdx.x * THREADS;

  // ---- Phase 0: stage embedding rows 0..63 into LDS (coalesced) ----
  for (int i = tid; i < VOCAB * H_DIM; i += THREADS) {
    const int r = i >> 6, c = i & 63;
    embed_s[r * E_STRIDE + c] = embed[r * H_DIM + c];
  }
  __syncthreads();

  // ---- Phase 1: per-block table precompute (tiny; weights come from L2) ----
  // ctxproj[v][j] = embed[v] @ gw1[0:64,j]   + gb1[j]
  // memproj[v][j] = embed[v] @ gw1[64:128,j]
  for (int i = tid; i < VOCAB * G1; i += THREADS) {
    const int v = i >> 5, j = i & 31;
    float ac = gb1[j], am = 0.f;
    for (int k = 0; k < H_DIM; ++k) {
      const float e = embed_s[v * E_STRIDE + k];
      ac += e * gw1[k * G1 + j];
      am += e * gw1[(H_DIM + k) * G1 + j];
    }
    ctxp_s[v * CP_STRIDE + j] = ac;
    memp_s[v * CP_STRIDE + j] = am;
  }
  // qproj[v][n] = embed[v] @ rw1[0:64,n] + rb1[n]   (v up to 65 -> read global)
  for (int i = tid; i < NVOCAB * H_DIM; i += THREADS) {
    const int v = i >> 6, n = i & 63;
    float a = rb1[n];
    for (int k = 0; k < H_DIM; ++k)
      a += embed[v * H_DIM + k] * rw1[k * H_DIM + n];
    qproj_s[v * H_DIM + n] = a;
  }
  __syncthreads();

  // ---- Phase 2: gating recurrence, one lane = one batch row ----
  {
    const int b = blockBase + tid;
    __builtin_prefetch(&seqs[b * SEQ_LEN], 0, 3);

    float4 w2[8];                        // gw2 in registers (broadcast load)
    #pragma unroll
    for (int jj = 0; jj < 8; ++jj) w2[jj] = ((const float4*)gw2)[jj];

    int st[SLOTS];                       // slot token ids (the entire mem state)
    #pragma unroll
    for (int t = 0; t < SLOTS; ++t)      // t < 8: idx forced to t, no scoring needed
      st[t] = seqs[b * SEQ_LEN + t];

    for (int t = SLOTS; t < T_STEPS; ++t) {
      const int tok = seqs[b * SEQ_LEN + t];
      float4 ctx[8];
      #pragma unroll
      for (int jj = 0; jj < 8; ++jj)
        ctx[jj] = *(const float4*)&ctxp_s[tok * CP_STRIDE + jj * 4];

      float best = -3.4e38f; int bi = 0;
      #pragma unroll
      for (int s = 0; s < SLOTS; ++s) {
        float z = 0.f;                   // = (h @ gw2); sigmoid/gb2 dropped (monotone)
        #pragma unroll
        for (int jj = 0; jj < 8; ++jj) {
          const float4 m = *(const float4*)&memp_s[st[s] * CP_STRIDE + jj * 4];
          const float4 c = ctx[jj], w = w2[jj];
          z += fmaxf(c.x + m.x, 0.f) * w.x + fmaxf(c.y + m.y, 0.f) * w.y
             + fmaxf(c.z + m.z, 0.f) * w.z + fmaxf(c.w + m.w, 0.f) * w.w;
        }
        if (z > best) { best = z; bi = s; }   // first-max semantics == jnp.argmax
      }
      #pragma unroll
      for (int s = 0; s < SLOTS; ++s) st[s] = (s == bi) ? tok : st[s];
    }
    #pragma unroll
    for (int s = 0; s < SLOTS; ++s) slot_s[tid][s] = (unsigned char)st[s];
  }
  __syncthreads();

  // ---- Phase 3: WMMA readout. Each wave handles two 16-batch tiles. ----
  const int wave  = tid >> 5;
  const int lane  = tid & 31;
  const int lm    = lane & 15;
  const int khalf = (lane >> 4) << 1;          // A/B K offset: lanes 16-31 hold K+2,K+3
  const int mhalf = (lane >> 4) << 3;          // C/D M offset: lanes 16-31 hold M+8
  float* hb = &h_s[wave * 16 * E_STRIDE];

  for (int tl = 0; tl < 2; ++tl) {
    const int rowBase = (wave * 2 + tl) * 16;  // batch row within block

    // Build A1 = summary tile (16x64) directly in WMMA A layout from slot ids:
    // summary[m][k] = 0.125 * sum_s embed[slot_tok[m][s]][k]
    v2f a1[16];
    #pragma unroll
    for (int c = 0; c < 16; ++c) {
      const int k = 4 * c + khalf;
      const unsigned char* sl = slot_s[rowBase + lm];
      float s0 = 0.f, s1 = 0.f;
      #pragma unroll
      for (int s = 0; s < SLOTS; ++s) {
        const int tk = sl[s];
        s0 += embed_s[tk * E_STRIDE + k];
        s1 += embed_s[tk * E_STRIDE + k + 1];
      }
      a1[c][0] = 0.125f * s0;
      a1[c][1] = 0.125f * s1;
    }

    // GEMM1: h = relu(qproj[qtok] + summary @ rw1_bot), written to LDS h tile
    for (int n0 = 0; n0 < H_DIM; n0 += 16) {
      v8f acc = {};
      #pragma unroll
      for (int c = 0; c < 16; ++c) {
        const int k = 4 * c + khalf;
        v2f bb;
        bb[0] = rw1[(H_DIM + k)     * H_DIM + n0 + lm];
        bb[1] = rw1[(H_DIM + k + 1) * H_DIM + n0 + lm];
        acc = wmma_f32_16x16x4(a1[c], bb, acc);
      }
      #pragma unroll
      for (int v = 0; v < 8; ++v) {
        const int m  = v + mhalf;
        const int qt = qtok[blockBase + rowBase + m];
        const float hv = acc[v] + qproj_s[qt * H_DIM + n0 + lm];
        hb[m * E_STRIDE + n0 + lm] = fmaxf(hv, 0.f);
      }
    }
    wave_fence();

    // Re-layout h for A operand of GEMM2 (via per-wave LDS tile)
    v2f a2[16];
    #pragma unroll
    for (int c = 0; c < 16; ++c) {
      const int k = 4 * c + khalf;
      a2[c][0] = hb[lm * E_STRIDE + k];
      a2[c][1] = hb[lm * E_STRIDE + k + 1];
    }

    // GEMM2: logits = h @ rw2 + rb2 -> global out
    const float bias2 = rb2[0];
    for (int n0 = 0; n0 < H_DIM; n0 += 16) {
      v8f acc = {};
      #pragma unroll
      for (int c = 0; c < 16; ++c) {
        const int k = 4 * c + khalf;
        v2f bb;
        bb[0] = rw2[k       * H_DIM + n0 + lm];
        bb[1] = rw2[(k + 1) * H_DIM + n0 + lm];
        acc = wmma_f32_16x16x4(a2[c], bb, acc);
      }
      #pragma unroll
      for (int v = 0; v < 8; ++v) {
        const int m = v + mhalf;
        out[(size_t)(blockBase + rowBase + m) * H_DIM + n0 + lm] = acc[v] + bias2;
      }
    }
    wave_fence();   // hb reused by next tile
  }
}

extern "C" void kernel_launch(void* const* d_in, const int* in_sizes, int n_in,
                              void* d_out, int out_size, void* d_ws, size_t ws_size,
                              hipStream_t stream) {
  (void)in_sizes; (void)n_in; (void)out_size; (void)d_ws; (void)ws_size;
  const int*   seqs  = (const int*)  d_in[0];
  const int*   qtk   = (const int*)  d_in[1];
  const float* embed = (const float*)d_in[2];
  const float* gw1   = (const float*)d_in[3];
  const float* gb1   = (const float*)d_in[4];
  const float* gw2   = (const float*)d_in[5];
  // d_in[6] = gb2: unused (constant shift before monotone sigmoid -> argmax invariant)
  const float* rw1   = (const float*)d_in[7];
  const float* rb1   = (const float*)d_in[8];
  const float* rw2   = (const float*)d_in[9];
  const float* rb2   = (const float*)d_in[10];
  float* outp = (float*)d_out;

  selective_model_kernel<<<B_TOTAL / THREADS, THREADS, 0, stream>>>(
      seqs, qtk, embed, gw1, gb1, gw2, rw1, rb1, rw2, rb2, outp);
}